// LightGCN_79044578115987
// MI455X (gfx1250) — compile-verified
//
#include <hip/hip_runtime.h>
#include <hip/hip_bf16.h>

// ---------------------------------------------------------------------------
// LightGCN propagation on MI455X (gfx1250).
//   ego0 = concat(user, item, brand)            (151000 x 64 f32, 38.7 MB)
//   for l in 0..2: ego_{l+1} = SpMM(A, ego_l); acc += ego_{l+1}
//   out  = acc / 4   (done via v_wmma_f32_16x16x4_f32, A = 0.25*I)
// Scatter-bound: 1.92G f32 L2 atomics dominate; embeddings are L2-resident.
// ---------------------------------------------------------------------------

#define NUM_USERS  100000
#define NUM_ITEMS  50000
#define NUM_BRANDS 1000
#define N_NODES    (NUM_USERS + NUM_ITEMS + NUM_BRANDS)   // 151000
#define EMBED_DIM  64
#define N_LAYERS   3
#define NODE_FLOATS ((size_t)N_NODES * EMBED_DIM)         // 9,664,000

typedef __attribute__((ext_vector_type(2))) float v2f;
typedef __attribute__((ext_vector_type(8))) float v8f;

// ---------------------------------------------------------------------------
// Kernel 1: build ego0 in workspace, seed acc (= d_out front), write echoes.
// One thread per float4 of the node matrix (16 float4 per node row).
// ---------------------------------------------------------------------------
__global__ void lgcn_init(const float* __restrict__ user,
                          const float* __restrict__ item,
                          const float* __restrict__ brand,
                          float* __restrict__ ego,
                          float* __restrict__ out) {
    int idx4 = blockIdx.x * blockDim.x + threadIdx.x;
    if (idx4 >= N_NODES * (EMBED_DIM / 4)) return;
    int node = idx4 >> 4;          // 16 float4 per row
    int comp = idx4 & 15;

    float4 v;
    if (node < NUM_USERS) {
        v = ((const float4*)user)[(size_t)node * 16 + comp];
    } else if (node < NUM_USERS + NUM_ITEMS) {
        v = ((const float4*)item)[(size_t)(node - NUM_USERS) * 16 + comp];
    } else {
        v = ((const float4*)brand)[(size_t)(node - NUM_USERS - NUM_ITEMS) * 16 + comp];
    }

    ((float4*)ego)[idx4] = v;   // ego_0
    ((float4*)out)[idx4] = v;   // acc seed

    // Echo outputs: d_out tail = user_emb then item_emb (same node order).
    if (node < NUM_USERS + NUM_ITEMS) {
        ((float4*)(out + NODE_FLOATS))[idx4] = v;
    }
}

// ---------------------------------------------------------------------------
// Kernel 2: zero the scatter destination (atomics need a clean base).
// ---------------------------------------------------------------------------
__global__ void lgcn_zero(float4* __restrict__ buf, int n4) {
    int i = blockIdx.x * blockDim.x + threadIdx.x;
    if (i < n4) buf[i] = make_float4(0.f, 0.f, 0.f, 0.f);
}

// ---------------------------------------------------------------------------
// Kernel 3: COO SpMM scatter. 16 lanes per edge, one float4 per lane.
// Gather hits L2 (38.7 MB working set); scatter = 4 relaxed agent-scope
// f32 atomics per lane (no-return -> STOREcnt path, resolved at L2).
// ---------------------------------------------------------------------------
__global__ void lgcn_spmm(const int*   __restrict__ rows,
                          const int*   __restrict__ cols,
                          const float* __restrict__ vals,
                          const float* __restrict__ xin,
                          float*       __restrict__ xout,
                          int n_edges) {
    int gid = blockIdx.x * blockDim.x + threadIdx.x;
    int e   = gid >> 4;
    if (e >= n_edges) return;
    int sub = gid & 15;

    int   r = rows[e];
    int   c = cols[e];
    float v = vals[e];

    float4 x = ((const float4*)(xin + (size_t)c * EMBED_DIM))[sub];
    float* dst = xout + (size_t)r * EMBED_DIM + sub * 4;

    __hip_atomic_fetch_add(dst + 0, v * x.x, __ATOMIC_RELAXED, __HIP_MEMORY_SCOPE_AGENT);
    __hip_atomic_fetch_add(dst + 1, v * x.y, __ATOMIC_RELAXED, __HIP_MEMORY_SCOPE_AGENT);
    __hip_atomic_fetch_add(dst + 2, v * x.z, __ATOMIC_RELAXED, __HIP_MEMORY_SCOPE_AGENT);
    __hip_atomic_fetch_add(dst + 3, v * x.w, __ATOMIC_RELAXED, __HIP_MEMORY_SCOPE_AGENT);
}

// ---------------------------------------------------------------------------
// Kernel 4: acc += ego_layer (elementwise float4).
// ---------------------------------------------------------------------------
__global__ void lgcn_accadd(float4* __restrict__ acc,
                            const float4* __restrict__ ego, int n4) {
    int i = blockIdx.x * blockDim.x + threadIdx.x;
    if (i >= n4) return;
    float4 a = acc[i], b = ego[i];
    a.x += b.x; a.y += b.y; a.z += b.z; a.w += b.w;
    acc[i] = a;
}

// ---------------------------------------------------------------------------
// Kernel 5: acc *= 0.25 via the matrix pipe.
// One wave per 64-float tile. Tile is loaded as B (4x16, row-major flat
// = k*16 + n, striped lanes 0-15:k0/k1, lanes 16-31:k2/k3 per ISA layout).
// A = 0.25 * I (16x4) built in registers (lane m = lane&15, k = (lane>>4)*2+v).
// D[m][n] = 0.25 * B[m][n] for m in 0..3 -> D VGPRs 0..3, lanes 0..15.
// ---------------------------------------------------------------------------
__global__ void lgcn_scale_wmma(float* __restrict__ buf, int ntiles) {
    int wave = blockIdx.x * (blockDim.x >> 5) + (threadIdx.x >> 5);
    if (wave >= ntiles) return;             // uniform per wave -> EXEC stays full
    int lane = threadIdx.x & 31;

    float* tile = buf + (size_t)wave * 64;

    int half = lane >> 4;                   // which K pair this lane holds
    int n    = lane & 15;

    // B operand = the data tile (4x16, flat index k*16 + n)
    v2f B;
    B.x = tile[(half * 2 + 0) * 16 + n];
    B.y = tile[(half * 2 + 1) * 16 + n];

    // A operand = 0.25 * I_{16x4}: A[m][k] = 0.25 iff m == k
    int m = lane & 15;
    v2f A;
    A.x = (m == (half * 2 + 0)) ? 0.25f : 0.0f;
    A.y = (m == (half * 2 + 1)) ? 0.25f : 0.0f;

    v8f C = {};
    v8f D = __builtin_amdgcn_wmma_f32_16x16x4_f32(
        /*neg_a=*/false, A, /*neg_b=*/false, B,
        /*c_mod=*/(short)0, C, /*reuse_a=*/false, /*reuse_b=*/false);

    // Useful result: rows m=0..3 live in D components 0..3, lanes 0..15.
    if (lane < 16) {
        tile[0 * 16 + lane] = D[0];
        tile[1 * 16 + lane] = D[1];
        tile[2 * 16 + lane] = D[2];
        tile[3 * 16 + lane] = D[3];
    }
}

// ---------------------------------------------------------------------------
// Host-side orchestration (graph-capture safe: only kernel launches).
// ---------------------------------------------------------------------------
extern "C" void kernel_launch(void* const* d_in, const int* in_sizes, int n_in,
                              void* d_out, int out_size, void* d_ws, size_t ws_size,
                              hipStream_t stream) {
    const int*   adj_rows = (const int*)  d_in[0];
    const int*   adj_cols = (const int*)  d_in[1];
    const float* adj_vals = (const float*)d_in[2];
    const float* user_emb = (const float*)d_in[3];
    const float* item_emb = (const float*)d_in[4];
    const float* brand_emb= (const float*)d_in[5];
    float* out = (float*)d_out;

    const int n_edges = in_sizes[0];

    // Workspace: two ping-pong node matrices (38.7 MB each).
    float* ego_a = (float*)d_ws;
    float* ego_b = ego_a + NODE_FLOATS;

    const int n4 = (int)(NODE_FLOATS / 4);            // float4 count: 2,416,000
    const int threads = 256;
    const int blk_node = (n4 + threads - 1) / threads;
    const int blk_edge = (int)(((long long)n_edges * 16 + threads - 1) / threads);
    const int blk_wmma = (N_NODES + (threads / 32) - 1) / (threads / 32);

    // 1) ego0 = concat; acc = ego0; echo user/item embeddings.
    lgcn_init<<<blk_node, threads, 0, stream>>>(user_emb, item_emb, brand_emb,
                                                ego_a, out);

    // 2) three propagation layers (ping-pong ego buffers).
    float* cur = ego_a;
    float* nxt = ego_b;
    for (int layer = 0; layer < N_LAYERS; ++layer) {
        lgcn_zero<<<blk_node, threads, 0, stream>>>((float4*)nxt, n4);
        lgcn_spmm<<<blk_edge, threads, 0, stream>>>(adj_rows, adj_cols, adj_vals,
                                                    cur, nxt, n_edges);
        lgcn_accadd<<<blk_node, threads, 0, stream>>>((float4*)out,
                                                      (const float4*)nxt, n4);
        float* t = cur; cur = nxt; nxt = t;
    }

    // 3) acc /= 4 through the WMMA pipe (one wave per 64-float node row).
    lgcn_scale_wmma<<<blk_wmma, threads, 0, stream>>>(out, N_NODES);
}